// LocalRegionMulti_11364483465331
// MI455X (gfx1250) — compile-verified
//
#include <hip/hip_runtime.h>

typedef __attribute__((ext_vector_type(2))) float v2f;
typedef __attribute__((ext_vector_type(8))) float v8f;

#define NB   16
#define NG   64
#define NK   12
#define NOUT 1024
#define BN_EPS 1e-5f

// ---------------------------------------------------------------------------
// Fold eval-mode BN + conv bias into per-channel scale/shift.
// scale = g * rsqrt(v + eps)   (g,v > 0  => scale > 0, so ReLU/affine commute
// with the max-over-K reduction)
// shift = (bW - mu) * scale + be
// Layout: op 0..3 = s-branches, op 4..7 = t-branches.  [8][1024] each.
// ---------------------------------------------------------------------------
__global__ __launch_bounds__(256) void prep_bn_kernel(
    const float* __restrict__ bWs, const float* __restrict__ gs,
    const float* __restrict__ bes, const float* __restrict__ mus,
    const float* __restrict__ vs,
    const float* __restrict__ bWt, const float* __restrict__ gt,
    const float* __restrict__ bet, const float* __restrict__ mut,
    const float* __restrict__ vt,
    float* __restrict__ scaleArr, float* __restrict__ shiftArr)
{
    int n = blockIdx.x * blockDim.x + threadIdx.x;   // 0..8191
    if (n >= 8 * NOUT) return;
    int op = n >> 10, ch = n & (NOUT - 1);
    const float *bW, *g, *be, *mu, *v;
    if (op < 4) {
        bW = bWs + op * NOUT; g = gs + op * NOUT; be = bes + op * NOUT;
        mu = mus + op * NOUT; v = vs + op * NOUT;
    } else {
        int i = op - 4;
        bW = bWt + i * NOUT; g = gt + i * NOUT; be = bet + i * NOUT;
        mu = mut + i * NOUT; v = vt + i * NOUT;
    }
    float sc = g[ch] * rsqrtf(v[ch] + BN_EPS);
    scaleArr[n] = sc;
    shiftArr[n] = (bW[ch] - mu[ch]) * sc + be[ch];
}

// ---------------------------------------------------------------------------
// Furthest point sampling on xt3 [16,256,3]; start index 0; 64 samples.
// One block per batch, one thread per point. Writes new_xyz [16,64,3].
// Argmax tie-break: lowest index (matches jnp.argmax).
// ---------------------------------------------------------------------------
__global__ __launch_bounds__(256) void fps_kernel(
    const float* __restrict__ xyz, float* __restrict__ new_xyz)
{
    __shared__ float sx[256], sy[256], sz[256], sd[256];
    __shared__ float rd[256];
    __shared__ int   ri[256];
    __shared__ int   s_last;

    const int b = blockIdx.x, t = threadIdx.x;
    const float* p = xyz + ((size_t)b * 256 + t) * 3;
    float px = p[0], py = p[1], pz = p[2];
    sx[t] = px; sy[t] = py; sz[t] = pz; sd[t] = 1e10f;
    if (t == 0) {
        s_last = 0;
        float* o = new_xyz + (size_t)b * NG * 3;
        o[0] = px; o[1] = py; o[2] = pz;
    }
    __syncthreads();

    for (int i = 1; i < NG; ++i) {
        int last = s_last;
        float dx = px - sx[last], dy = py - sy[last], dz = pz - sz[last];
        float d = dx * dx + dy * dy + dz * dz;
        float mind = fminf(sd[t], d);
        sd[t] = mind;
        rd[t] = mind; ri[t] = t;
        __syncthreads();
        for (int s = 128; s > 0; s >>= 1) {
            if (t < s) {
                float od = rd[t + s]; int oi = ri[t + s];
                if (od > rd[t] || (od == rd[t] && oi < ri[t])) { rd[t] = od; ri[t] = oi; }
            }
            __syncthreads();
        }
        if (t == 0) {
            int nxt = ri[0];
            s_last = nxt;
            float* o = new_xyz + ((size_t)b * NG + i) * 3;
            o[0] = sx[nxt]; o[1] = sy[nxt]; o[2] = sz[nxt];
        }
        __syncthreads();
    }
}

// ---------------------------------------------------------------------------
// kNN: one wave per query; 8 point sets (grid.y): 0..3 = xs0..3, 4..7 = xt0..3.
// Each lane keeps a register-resident sorted top-12 (unrolled bubble insert),
// then lane 0 merges the 32 sorted lists via LDS.
// knn_out layout: [set][B*G][12] ints.
// ---------------------------------------------------------------------------
__global__ __launch_bounds__(256) void knn_kernel(
    const float* __restrict__ xs0, const float* __restrict__ xs1,
    const float* __restrict__ xs2, const float* __restrict__ xs3,
    const float* __restrict__ xt0, const float* __restrict__ xt1,
    const float* __restrict__ xt2, const float* __restrict__ xt3,
    const float* __restrict__ new_xyz, int* __restrict__ knn_out)
{
    __shared__ float md[8][32][NK];
    __shared__ int   mi[8][32][NK];

    const int set = blockIdx.y;
    const float* xyz; int N;
    switch (set) {
        case 0:  xyz = xs0; N = 2048; break;
        case 1:  xyz = xs1; N = 1024; break;
        case 2:  xyz = xs2; N = 512;  break;
        case 3:  xyz = xs3; N = 256;  break;
        case 4:  xyz = xt0; N = 2048; break;
        case 5:  xyz = xt1; N = 1024; break;
        case 6:  xyz = xt2; N = 512;  break;
        default: xyz = xt3; N = 256;  break;
    }

    const int wave = threadIdx.x >> 5, lane = threadIdx.x & 31;
    const int q = blockIdx.x * 8 + wave;          // 0..1023 == b*64+g
    const int b = q >> 6;

    const float qx = new_xyz[q * 3 + 0];
    const float qy = new_xyz[q * 3 + 1];
    const float qz = new_xyz[q * 3 + 2];

    float bd[NK]; int bi[NK];
#pragma unroll
    for (int j = 0; j < NK; ++j) { bd[j] = 3.0e38f; bi[j] = 0; }

    const float* base = xyz + (size_t)b * N * 3;
    for (int n = lane; n < N; n += 32) {
        float dx = base[n * 3 + 0] - qx;
        float dy = base[n * 3 + 1] - qy;
        float dz = base[n * 3 + 2] - qz;
        float d = dx * dx + dy * dy + dz * dz;
        if (d < bd[NK - 1]) {
            bd[NK - 1] = d; bi[NK - 1] = n;
#pragma unroll
            for (int j = NK - 1; j > 0; --j) {
                if (bd[j] < bd[j - 1]) {
                    float tf = bd[j]; bd[j] = bd[j - 1]; bd[j - 1] = tf;
                    int   ti = bi[j]; bi[j] = bi[j - 1]; bi[j - 1] = ti;
                }
            }
        }
    }
#pragma unroll
    for (int j = 0; j < NK; ++j) { md[wave][lane][j] = bd[j]; mi[wave][lane][j] = bi[j]; }
    __syncthreads();

    if (lane == 0) {
        float od[NK]; int oi[NK];
#pragma unroll
        for (int j = 0; j < NK; ++j) { od[j] = md[wave][0][j]; oi[j] = mi[wave][0][j]; }
        for (int l = 1; l < 32; ++l) {
            for (int j = 0; j < NK; ++j) {
                float d = md[wave][l][j];
                if (d >= od[NK - 1]) break;     // lane lists are sorted
                od[NK - 1] = d; oi[NK - 1] = mi[wave][l][j];
#pragma unroll
                for (int jj = NK - 1; jj > 0; --jj) {
                    if (od[jj] < od[jj - 1]) {
                        float tf = od[jj]; od[jj] = od[jj - 1]; od[jj - 1] = tf;
                        int   ti = oi[jj]; oi[jj] = oi[jj - 1]; oi[jj - 1] = ti;
                    }
                }
            }
        }
        int* o = knn_out + ((size_t)set * 1024 + q) * NK;
#pragma unroll
        for (int j = 0; j < NK; ++j) o[j] = oi[j];
    }
}

// ---------------------------------------------------------------------------
// Fused gather + GEMM(WMMA f32 16x16x4) + BN/ReLU + max-over-K.
// One block per PAIR of (b,g) groups (M = 2x16): every B fragment feeds two
// WMMAs (one per M-tile), halving L2 weight traffic vs one group per block.
// Each A fragment (LDS, ds_load_b64) feeds four WMMAs (4 N-tiles).
// 8 waves x 4 N-tiles x 2 passes = 64 tiles = 1024 channels.
// D layout: lane L holds channel N = nt*16 + (L&15); lanes<16 rows 0..7,
// lanes>=16 rows 8..15 (valid rows <=11 => elements 0..3 only).
// ---------------------------------------------------------------------------
__device__ __forceinline__ void tile_epilogue(
    v8f c, int nt, int lrow, int lhi,
    const float* __restrict__ scaleArr, const float* __restrict__ shiftArr,
    float* __restrict__ outrow)
{
    float m03 = fmaxf(fmaxf(c[0], c[1]), fmaxf(c[2], c[3]));
    float m47 = fmaxf(fmaxf(c[4], c[5]), fmaxf(c[6], c[7]));
    float m = lhi ? m03 : fmaxf(m03, m47);       // mask pad rows 12..15
    m = fmaxf(m, __shfl_xor(m, 16, 32));         // combine row halves
    if (lhi == 0) {
        int n = nt * 16 + lrow;
        outrow[n] = fmaxf(m * scaleArr[n] + shiftArr[n], 0.0f);
    }
}

#define WMMA_F32(a, b, c) \
    __builtin_amdgcn_wmma_f32_16x16x4_f32(false, (a), false, (b), (short)0, (c), false, false)

template <int C>
__global__ __launch_bounds__(256) void gemm_graph_op(
    const float* __restrict__ feat, int N,
    const int* __restrict__ knn,
    const float* __restrict__ W,                 // [1024, C] row-major
    const float* __restrict__ scaleArr, const float* __restrict__ shiftArr,
    float* __restrict__ out)                     // [B*G, 1024]
{
    constexpr int CH  = (C > 256) ? 256 : C;     // LDS chunk of the K dim
    constexpr int STR = CH + 4;                  // +4 floats: conflict-free b64
    __shared__ float As[32 * STR];               // two 16-row M-tiles
    __shared__ int   sidx[2 * NK];

    const int bg0 = blockIdx.x * 2;              // two (b,g) groups per block
    const int tid = threadIdx.x;
    const int wave = tid >> 5, lane = tid & 31;
    const int lrow = lane & 15, lhi = lane >> 4;

    if (tid < 2 * NK) sidx[tid] = knn[bg0 * NK + tid];

    float* outrow0 = out + (size_t)bg0 * NOUT;
    float* outrow1 = outrow0 + NOUT;

    for (int pass = 0; pass < 2; ++pass) {
        const int nt0 = pass * 32 + wave * 4;    // 4 adjacent 16-channel tiles
        v8f c00 = {}, c01 = {}, c02 = {}, c03 = {};   // group 0
        v8f c10 = {}, c11 = {}, c12 = {}, c13 = {};   // group 1
        const float* w0 = W + (size_t)(nt0 * 16 + lrow) * C + 2 * lhi;

        for (int coff = 0; coff < C; coff += CH) {
            __syncthreads();                     // As reuse + sidx visibility
            for (int e = tid; e < 32 * CH; e += 256) {
                int row = e / CH, col = e - row * CH;
                int grp = row >> 4, r = row & 15;
                float vv = 0.0f;
                if (r < NK) {
                    int b = (bg0 + grp) >> 6;
                    vv = feat[((size_t)b * N + sidx[grp * NK + r]) * C + coff + col];
                }
                As[row * STR + col] = vv;
            }
            __syncthreads();

            const float* a0 = As + lrow * STR + 2 * lhi;
            const float* a1 = a0 + 16 * STR;
            const float* wk = w0 + coff;
#pragma unroll 2
            for (int k = 0; k < CH; k += 4) {
                v2f av0 = *(const v2f*)(a0 + k);
                v2f av1 = *(const v2f*)(a1 + k);
                v2f b0  = *(const v2f*)(wk + k);
                v2f b1  = *(const v2f*)(wk + 16 * C + k);
                v2f b2  = *(const v2f*)(wk + 32 * C + k);
                v2f b3  = *(const v2f*)(wk + 48 * C + k);
                c00 = WMMA_F32(av0, b0, c00);
                c10 = WMMA_F32(av1, b0, c10);
                c01 = WMMA_F32(av0, b1, c01);
                c11 = WMMA_F32(av1, b1, c11);
                c02 = WMMA_F32(av0, b2, c02);
                c12 = WMMA_F32(av1, b2, c12);
                c03 = WMMA_F32(av0, b3, c03);
                c13 = WMMA_F32(av1, b3, c13);
            }
        }
        tile_epilogue(c00, nt0 + 0, lrow, lhi, scaleArr, shiftArr, outrow0);
        tile_epilogue(c01, nt0 + 1, lrow, lhi, scaleArr, shiftArr, outrow0);
        tile_epilogue(c02, nt0 + 2, lrow, lhi, scaleArr, shiftArr, outrow0);
        tile_epilogue(c03, nt0 + 3, lrow, lhi, scaleArr, shiftArr, outrow0);
        tile_epilogue(c10, nt0 + 0, lrow, lhi, scaleArr, shiftArr, outrow1);
        tile_epilogue(c11, nt0 + 1, lrow, lhi, scaleArr, shiftArr, outrow1);
        tile_epilogue(c12, nt0 + 2, lrow, lhi, scaleArr, shiftArr, outrow1);
        tile_epilogue(c13, nt0 + 3, lrow, lhi, scaleArr, shiftArr, outrow1);
    }
}

// ---------------------------------------------------------------------------
extern "C" void kernel_launch(void* const* d_in, const int* in_sizes, int n_in,
                              void* d_out, int out_size, void* d_ws, size_t ws_size,
                              hipStream_t stream)
{
    const float* fs[4] = {(const float*)d_in[0], (const float*)d_in[4],
                          (const float*)d_in[8], (const float*)d_in[12]};
    const float* xs[4] = {(const float*)d_in[1], (const float*)d_in[5],
                          (const float*)d_in[9], (const float*)d_in[13]};
    const float* ft[4] = {(const float*)d_in[2], (const float*)d_in[6],
                          (const float*)d_in[10], (const float*)d_in[14]};
    const float* xt[4] = {(const float*)d_in[3], (const float*)d_in[7],
                          (const float*)d_in[11], (const float*)d_in[15]};
    const float* Ws    = (const float*)d_in[16];
    const float* bWs   = (const float*)d_in[17];
    const float* gs    = (const float*)d_in[18];
    const float* bes   = (const float*)d_in[19];
    const float* mus   = (const float*)d_in[20];
    const float* vs    = (const float*)d_in[21];
    const float* Wt0   = (const float*)d_in[22];
    const float* Wt123 = (const float*)d_in[23];
    const float* bWt   = (const float*)d_in[24];
    const float* gt    = (const float*)d_in[25];
    const float* bet   = (const float*)d_in[26];
    const float* mut   = (const float*)d_in[27];
    const float* vt    = (const float*)d_in[28];

    float* out = (float*)d_out;

    // Scratch layout (all regions fully rewritten every call; ~460 KB total)
    float* new_xyz  = (float*)d_ws;              // [16*64*3]
    int*   knn      = (int*)(new_xyz + NB * NG * 3);       // [8][1024][12]
    float* scaleArr = (float*)(knn + 8 * NB * NG * NK);    // [8][1024]
    float* shiftArr = scaleArr + 8 * NOUT;                 // [8][1024]

    static const int NSh[4] = {2048, 1024, 512, 256};
    const size_t OP_STRIDE = (size_t)NB * NG * NOUT;       // per-output offset
    const int    GEMM_GRID = (NB * NG) / 2;                // 2 groups per block

    prep_bn_kernel<<<32, 256, 0, stream>>>(bWs, gs, bes, mus, vs,
                                           bWt, gt, bet, mut, vt,
                                           scaleArr, shiftArr);
    fps_kernel<<<16, 256, 0, stream>>>(xt[3], new_xyz);
    knn_kernel<<<dim3(128, 8), 256, 0, stream>>>(xs[0], xs[1], xs[2], xs[3],
                                                 xt[0], xt[1], xt[2], xt[3],
                                                 new_xyz, knn);

    // out_s[i]  (ops 0..3): C=256, weights Ws[i]
    for (int i = 0; i < 4; ++i) {
        gemm_graph_op<256><<<GEMM_GRID, 256, 0, stream>>>(
            fs[i], NSh[i], knn + i * NB * NG * NK,
            Ws + (size_t)i * NOUT * 256,
            scaleArr + i * NOUT, shiftArr + i * NOUT,
            out + (size_t)i * OP_STRIDE);
    }
    // out_t[0] (op 4): C=1024, weights Wt0
    gemm_graph_op<1024><<<GEMM_GRID, 256, 0, stream>>>(
        ft[0], 2048, knn + 4 * NB * NG * NK,
        Wt0, scaleArr + 4 * NOUT, shiftArr + 4 * NOUT,
        out + (size_t)4 * OP_STRIDE);
    // out_t[1..3] (ops 5..7): C=256, weights Wt123[i-1]
    for (int i = 1; i < 4; ++i) {
        gemm_graph_op<256><<<GEMM_GRID, 256, 0, stream>>>(
            ft[i], NSh[i], knn + (size_t)(4 + i) * NB * NG * NK,
            Wt123 + (size_t)(i - 1) * NOUT * 256,
            scaleArr + (4 + i) * NOUT, shiftArr + (4 + i) * NOUT,
            out + (size_t)(4 + i) * OP_STRIDE);
    }
}